// GCNNet_63608465654163
// MI455X (gfx1250) — compile-verified
//
#include <hip/hip_runtime.h>

// -------------------------------------------------------------------------
// 2-layer GCN, fp32 throughout (matches reference numerics).
// Whole working set (~45 MB) fits MI455X's 192 MB L2 -> L2-resident problem.
// Dense transform uses V_WMMA_F32_16X16X4_F32 (exact fp32 matrix pipe);
// sparse aggregation uses L2 f32 atomics with self-loops folded into a
// non-atomic init pass.
// -------------------------------------------------------------------------

typedef float v2f __attribute__((ext_vector_type(2)));
typedef float v8f __attribute__((ext_vector_type(8)));

#define FEAT 64
#define OUTC 2

// deg starts at 1.0 (self-loop)
__global__ __launch_bounds__(256) void k_fill1(float* __restrict__ p, int n) {
  int i = blockIdx.x * blockDim.x + threadIdx.x;
  if (i < n) p[i] = 1.0f;
}

// deg[dst] += 1 for each real edge
__global__ __launch_bounds__(256) void k_deg(const int* __restrict__ dst,
                                             float* __restrict__ deg, int E, int N) {
  int e = blockIdx.x * blockDim.x + threadIdx.x;
  if (e < E) {
    int d = dst[e];
    if ((unsigned)d < (unsigned)N) atomicAdd(&deg[d], 1.0f);
  }
}

// dis = deg^{-1/2} in place
__global__ __launch_bounds__(256) void k_rsqrt(float* __restrict__ p, int n) {
  int i = blockIdx.x * blockDim.x + threadIdx.x;
  if (i < n) p[i] = rsqrtf(p[i]);
}

// Y[M,64] = X[M,64] @ W[64,64] via V_WMMA_F32_16X16X4_F32.
// One wave -> one 16x16 output tile; 4 waves per block cover N=0..63.
__global__ __launch_bounds__(128) void k_gemm1_wmma(const float* __restrict__ X,
                                                    const float* __restrict__ W,
                                                    float* __restrict__ Y) {
  const int lane = threadIdx.x & 31;
  const int wv   = threadIdx.x >> 5;   // N-tile select, 0..3
  const int m0   = blockIdx.x * 16;
  const int n0   = wv * 16;
  const int r    = lane & 15;
  const int hi   = lane >> 4;          // 0: K pair {0,1}, 1: K pair {2,3}

  const float* xrow = X + (size_t)(m0 + r) * FEAT;
  v8f c = {};
#pragma unroll
  for (int k0 = 0; k0 < FEAT; k0 += 4) {
    v2f a, b;
    // A 16x4 fragment: lanes 0-15 hold K={k0,k0+1}, lanes 16-31 K={k0+2,k0+3}
    a.x = xrow[k0 + 2 * hi + 0];
    a.y = xrow[k0 + 2 * hi + 1];
    // B 4x16 fragment: VGPR0 rows K=k0 / k0+2, VGPR1 rows K=k0+1 / k0+3
    b.x = W[(k0 + 2 * hi + 0) * FEAT + n0 + r];
    b.y = W[(k0 + 2 * hi + 1) * FEAT + n0 + r];
    c = __builtin_amdgcn_wmma_f32_16x16x4_f32(false, a, false, b,
                                              (short)0, c, false, false);
  }
  // C layout: VGPR v -> rows m0+v (lanes 0-15) and m0+v+8 (lanes 16-31)
#pragma unroll
  for (int v = 0; v < 8; ++v)
    Y[(size_t)(m0 + v + hi * 8) * FEAT + n0 + r] = c[v];
}

// h = dis[i]^2 * hw1[i]  (self-loop contribution; fully initializes h)
__global__ __launch_bounds__(256) void k_selfinit(const float* __restrict__ hw,
                                                  const float* __restrict__ dis,
                                                  float* __restrict__ h, int N) {
  int idx = blockIdx.x * blockDim.x + threadIdx.x;
  if (idx >= N * FEAT) return;
  int i = idx >> 6;
  float w = dis[i];
  h[idx] = w * w * hw[idx];
}

// scatter-add over real edges, 64 threads per edge (edge wave-uniform)
__global__ __launch_bounds__(256) void k_edge1(const int* __restrict__ src,
                                               const int* __restrict__ dst,
                                               const float* __restrict__ dis,
                                               const float* __restrict__ hw,
                                               float* __restrict__ h, int E, int N) {
  int idx = blockIdx.x * blockDim.x + threadIdx.x;
  if (idx >= E * FEAT) return;
  int e = idx >> 6;          // uniform within a wave32
  int f = idx & 63;
  int s = src[e];
  int d = dst[e];
  s = __builtin_amdgcn_readfirstlane(s);   // scalarize wave-uniform indices
  d = __builtin_amdgcn_readfirstlane(d);
  if ((unsigned)s >= (unsigned)N || (unsigned)d >= (unsigned)N) return;
  float w = dis[s] * dis[d];
  atomicAdd(&h[(size_t)d * FEAT + f], w * hw[(size_t)s * FEAT + f]);
}

// h = relu(h + b1)
__global__ __launch_bounds__(256) void k_biasrelu(float* __restrict__ h,
                                                  const float* __restrict__ b1, int N) {
  int idx = blockIdx.x * blockDim.x + threadIdx.x;
  if (idx >= N * FEAT) return;
  int f = idx & 63;
  h[idx] = fmaxf(h[idx] + b1[f], 0.0f);
}

// hw2[N,2] = h[N,64] @ W2[64,2]  (N too narrow for WMMA to pay off)
__global__ __launch_bounds__(256) void k_gemm2(const float* __restrict__ H,
                                               const float* __restrict__ W2,
                                               float* __restrict__ Y, int N) {
  int idx = blockIdx.x * blockDim.x + threadIdx.x;
  if (idx >= N * OUTC) return;
  int i = idx >> 1, c = idx & 1;
  const float* hr = H + (size_t)i * FEAT;
  float acc = 0.0f;
#pragma unroll
  for (int k = 0; k < FEAT; ++k) acc += hr[k] * W2[k * OUTC + c];
  Y[idx] = acc;
}

// out = dis[i]^2 * hw2[i] + b2  (self-loop + bias; fully initializes out)
__global__ __launch_bounds__(256) void k_outinit(const float* __restrict__ hw2,
                                                 const float* __restrict__ dis,
                                                 const float* __restrict__ b2,
                                                 float* __restrict__ out, int N) {
  int idx = blockIdx.x * blockDim.x + threadIdx.x;
  if (idx >= N * OUTC) return;
  int i = idx >> 1, c = idx & 1;
  float w = dis[i];
  out[idx] = w * w * hw2[idx] + b2[c];
}

__global__ __launch_bounds__(256) void k_edge2(const int* __restrict__ src,
                                               const int* __restrict__ dst,
                                               const float* __restrict__ dis,
                                               const float* __restrict__ hw2,
                                               float* __restrict__ out, int E, int N) {
  int idx = blockIdx.x * blockDim.x + threadIdx.x;
  if (idx >= E * OUTC) return;
  int e = idx >> 1, c = idx & 1;
  int s = src[e];
  int d = dst[e];
  if ((unsigned)s >= (unsigned)N || (unsigned)d >= (unsigned)N) return;
  float w = dis[s] * dis[d];
  atomicAdd(&out[(size_t)d * OUTC + c], w * hw2[(size_t)s * OUTC + c]);
}

extern "C" void kernel_launch(void* const* d_in, const int* in_sizes, int n_in,
                              void* d_out, int out_size, void* d_ws, size_t ws_size,
                              hipStream_t stream) {
  const float* x  = (const float*)d_in[0];
  const int*   ei = (const int*)d_in[1];   // [2, E] row-major
  const float* W1 = (const float*)d_in[2];
  const float* b1 = (const float*)d_in[3];
  const float* W2 = (const float*)d_in[4];
  const float* b2 = (const float*)d_in[5];
  float* out = (float*)d_out;

  const int N = in_sizes[0] / FEAT;   // 50000
  const int E = in_sizes[1] / 2;      // 800000
  const int* src = ei;
  const int* dst = ei + E;

  // workspace carve-out (~26.2 MB total, 256B aligned slabs)
  char* ws = (char*)d_ws;
  auto al = [](size_t v) { return (v + 255) & ~(size_t)255; };
  size_t off = 0;
  float* dis = (float*)(ws + off); off += al((size_t)N * 4);
  float* hw1 = (float*)(ws + off); off += al((size_t)N * FEAT * 4);
  float* h   = (float*)(ws + off); off += al((size_t)N * FEAT * 4);
  float* hw2 = (float*)(ws + off); off += al((size_t)N * OUTC * 4);
  (void)off; (void)ws_size; (void)n_in; (void)out_size;

  const int T = 256;
  const int bN   = (N + T - 1) / T;
  const int bE   = (E + T - 1) / T;
  const int bNF  = (N * FEAT + T - 1) / T;
  const int bEF  = (E * FEAT + T - 1) / T;
  const int bN2  = (N * OUTC + T - 1) / T;
  const int bE2  = (E * OUTC + T - 1) / T;

  // normalization: deg -> dis = deg^{-1/2}
  k_fill1 <<<bN, T, 0, stream>>>(dis, N);
  k_deg   <<<bE, T, 0, stream>>>(dst, dis, E, N);
  k_rsqrt <<<bN, T, 0, stream>>>(dis, N);

  // layer 1
  k_gemm1_wmma<<<N / 16, 128, 0, stream>>>(x, W1, hw1);
  k_selfinit  <<<bNF, T, 0, stream>>>(hw1, dis, h, N);
  k_edge1     <<<bEF, T, 0, stream>>>(src, dst, dis, hw1, h, E, N);
  k_biasrelu  <<<bNF, T, 0, stream>>>(h, b1, N);

  // layer 2
  k_gemm2  <<<bN2, T, 0, stream>>>(h, W2, hw2, N);
  k_outinit<<<bN2, T, 0, stream>>>(hw2, dis, b2, out, N);
  k_edge2  <<<bE2, T, 0, stream>>>(src, dst, dis, hw2, out, E, N);
}